// CatEdgeGraphLayer_33277406609831
// MI455X (gfx1250) — compile-verified
//
#include <hip/hip_runtime.h>

// Problem constants (reference: B=128, N=64, D=128, OUT=128)
#define BATCH 128
#define NAG   64
#define DFEAT 128
#define NOUT  128
#define WROW  258   // 2*D + 2

typedef __attribute__((ext_vector_type(2))) float v2f;
typedef __attribute__((ext_vector_type(8))) float v8f;

// out[b,i,o] = relu( 63*(W1 f_i + bias)[o] + (W2 S1[b,i])[o] + (W3 s3[b,i])[o] )
//   S1[b,i]  = sum_{j != i} A[b,i,j] * f_j
//   s3[b,i]  = sum_{j != i} A[b,i,j] * diff[b,i,j,:]
__launch_bounds__(256, 2)
__global__ void cat_edge_graph_kernel(const float* __restrict__ diff,   // [B,N,N,2]
                                      const float* __restrict__ feat,   // [B,N,D]
                                      const float* __restrict__ A,      // [B,N,N]
                                      const float* __restrict__ W,      // [OUT, 2D+2]
                                      const float* __restrict__ bias,   // [OUT]
                                      float* __restrict__ out)          // [B,N,OUT]
{
    __shared__ float sF[NAG * DFEAT];   // 32 KB  agent features, row-major [i][d]
    __shared__ float sA[NAG * NAG];     // 16 KB  adjacency, diagonal zeroed
    __shared__ float sS1[NAG * DFEAT];  // 32 KB  S1 = Aod @ F
    __shared__ float sS3[NAG * 2];      // 512 B  weighted diff sums

    const int b    = blockIdx.x;
    const int tid  = threadIdx.x;
    const int lane = tid & 31;          // wave32
    const int wave = tid >> 5;          // 8 waves
    const int half = lane >> 4;         // lane group 0/1 (K split in WMMA frags)
    const int l16  = lane & 15;

    const float* Fb = feat + (size_t)b * NAG * DFEAT;
    const float* Ab = A    + (size_t)b * NAG * NAG;
    const float* Db = diff + (size_t)b * NAG * NAG * 2;

    // ---- Phase 0: stage F and diag-zeroed A into LDS ----
    for (int idx = tid; idx < NAG * DFEAT / 4; idx += 256)
        ((float4*)sF)[idx] = ((const float4*)Fb)[idx];
    for (int idx = tid; idx < NAG * NAG; idx += 256) {
        int i = idx >> 6, j = idx & 63;
        sA[idx] = (i == j) ? 0.0f : Ab[idx];
    }
    __syncthreads();

    // s3[i][k] = sum_j Aod[i][j] * diff[i][j][k]  (128 reductions of length 64)
    if (tid < NAG * 2) {
        const int i = tid >> 1, k = tid & 1;
        float s = 0.0f;
        for (int j = 0; j < NAG; ++j)
            s += sA[i * NAG + j] * Db[(i * NAG + j) * 2 + k];
        sS3[i * 2 + k] = s;
    }

    // ---- Phase 1: S1 = Aod @ F   (M=64, K=64, N=128) ----
    // wave w -> 16-wide column strip w; 4 M-tiles per wave; K loop of 16 x (16x16x4)
    {
        const int n = wave * 16 + l16;        // column of F / S1
        v8f acc[4] = {};
        for (int k0 = 0; k0 < NAG; k0 += 4) {
            const int ka = k0 + 2 * half;     // this half-wave's K pair
            v2f bfrag;
            bfrag.x = sF[(ka + 0) * DFEAT + n];
            bfrag.y = sF[(ka + 1) * DFEAT + n];
#pragma unroll
            for (int mi = 0; mi < 4; ++mi) {
                const int m = mi * 16 + l16;
                v2f afrag = *(const v2f*)&sA[m * NAG + ka];
                acc[mi] = __builtin_amdgcn_wmma_f32_16x16x4_f32(
                    false, afrag, false, bfrag, (short)0, acc[mi], false, false);
            }
        }
#pragma unroll
        for (int mi = 0; mi < 4; ++mi)
#pragma unroll
            for (int r = 0; r < 8; ++r) {
                const int m = mi * 16 + r + 8 * half;
                sS1[m * DFEAT + n] = acc[mi][r];
            }
    }
    __syncthreads();

    // ---- Phase 2: OUT = 63*F @ W1^T + S1 @ W2^T + s3 @ W3^T + 63*bias, ReLU ----
    // wave w -> output columns [16w, 16w+16); W fragments loaded once per K step,
    // reused by 4 M-tiles (B-operand reuse), 63x scale folded into the W1 fragment.
    {
        const int o = wave * 16 + l16;
        const float* Wo = W + o * WROW;
        const float w3a = Wo[256], w3b = Wo[257];
        const float bs  = 63.0f * bias[o];

        v8f acc[4];
#pragma unroll
        for (int mi = 0; mi < 4; ++mi)
#pragma unroll
            for (int r = 0; r < 8; ++r) {
                const int m = mi * 16 + r + 8 * half;
                acc[mi][r] = bs + sS3[m * 2] * w3a + sS3[m * 2 + 1] * w3b;
            }

        for (int k0 = 0; k0 < DFEAT; k0 += 4) {
            const int ka = k0 + 2 * half;
            // part 1: (63*F) @ W1^T   — W1^T[k][o] = W[o][k]
            v2f bfrag = *(const v2f*)&Wo[ka];       // 8B aligned: o*258+ka is even
            bfrag.x *= 63.0f;
            bfrag.y *= 63.0f;
#pragma unroll
            for (int mi = 0; mi < 4; ++mi) {
                const int m = mi * 16 + l16;
                v2f afrag = *(const v2f*)&sF[m * DFEAT + ka];
                acc[mi] = __builtin_amdgcn_wmma_f32_16x16x4_f32(
                    false, afrag, false, bfrag, (short)0, acc[mi], false, false);
            }
            // part 2: S1 @ W2^T       — W2^T[k][o] = W[o][128+k]
            v2f bfrag2 = *(const v2f*)&Wo[DFEAT + ka];
#pragma unroll
            for (int mi = 0; mi < 4; ++mi) {
                const int m = mi * 16 + l16;
                v2f afrag = *(const v2f*)&sS1[m * DFEAT + ka];
                acc[mi] = __builtin_amdgcn_wmma_f32_16x16x4_f32(
                    false, afrag, false, bfrag2, (short)0, acc[mi], false, false);
            }
        }

        float* Ob = out + (size_t)b * NAG * NOUT;
#pragma unroll
        for (int mi = 0; mi < 4; ++mi)
#pragma unroll
            for (int r = 0; r < 8; ++r) {
                const int m = mi * 16 + r + 8 * half;
                const float v = acc[mi][r];
                Ob[m * NOUT + o] = v > 0.0f ? v : 0.0f;
            }
    }
}

extern "C" void kernel_launch(void* const* d_in, const int* in_sizes, int n_in,
                              void* d_out, int out_size, void* d_ws, size_t ws_size,
                              hipStream_t stream) {
    const float* diff = (const float*)d_in[0];  // [B,N,N,2]
    const float* feat = (const float*)d_in[1];  // [B,N,D]
    const float* A    = (const float*)d_in[2];  // [B,N,N]
    const float* W    = (const float*)d_in[3];  // [OUT, 2D+2]
    const float* bias = (const float*)d_in[4];  // [OUT]

    float* out = (float*)d_out;
    const size_t diff_elems = (size_t)BATCH * NAG * NAG * 2;  // 1,048,576

    // Output 0: diff_vecs passthrough (d2d copy, graph-capture safe)
    hipMemcpyAsync(out, diff, diff_elems * sizeof(float),
                   hipMemcpyDeviceToDevice, stream);

    // Output 1: relu(edge-aggregated linear) -> [B,N,OUT]
    cat_edge_graph_kernel<<<BATCH, 256, 0, stream>>>(
        diff, feat, A, W, bias, out + diff_elems);
}